// TextModel_76862734729223
// MI455X (gfx1250) — compile-verified
//
#include <hip/hip_runtime.h>
#include <hip/hip_bf16.h>

// Problem constants (from reference): B=16, S=2048, H=2048, N=64
#define BB   16
#define SS   2048
#define HH   2048
#define NID  64

#define NEV  128          // 2 events (beg,end) per id
#define NS   16           // s-chunks
#define CS   (SS / NS)    // 128 rows per chunk
#define RB   16           // rows per LDS buffer
#define NSTG (CS / RB)    // 8 pipeline stages per chunk
#define HT   256          // columns per kA block (one per thread)

#define HB   128          // columns per kB block (WMMA tile region)
#define MP   129          // padded row stride of M in LDS (bank-conflict-free A loads)

typedef __attribute__((ext_vector_type(2))) float v2f;
typedef __attribute__((ext_vector_type(8))) float v8f;

// ---------------------------------------------------------------------------
// CDNA5 async-to-LDS helpers (inline asm; portable across toolchains)
// ---------------------------------------------------------------------------
__device__ __forceinline__ void async_fill_rows(const float* gs /*row0 of this fill*/,
                                                unsigned lbase /*LDS byte addr of buffer*/,
                                                int t) {
  // Copy RB(=16) rows x 1KB = 16KB into LDS. 256 threads x 4 chunks x 16B.
#pragma unroll
  for (int i = 0; i < 4; ++i) {
    unsigned c    = (unsigned)t + 256u * (unsigned)i;
    unsigned row  = c >> 6;                 // 64 chunks per row
    unsigned offb = (c & 63u) * 16u;        // byte offset within row
    unsigned laddr = lbase + c * 16u;
    const char* g = (const char*)gs + (size_t)row * (HH * 4) + offb;
    asm volatile("global_load_async_to_lds_b128 %0, %1, off"
                 :: "v"(laddr), "v"(g) : "memory");
  }
}

__device__ __forceinline__ void wait_async_le4() {
  asm volatile("s_wait_asynccnt 0x4" ::: "memory");
}
__device__ __forceinline__ void wait_async_0() {
  asm volatile("s_wait_asynccnt 0x0" ::: "memory");
}

// ---------------------------------------------------------------------------
// Kernel 0: per-batch span extraction + sorted event list + pad_mask output
// ---------------------------------------------------------------------------
__global__ __launch_bounds__(256)
void k0_spans(const int* __restrict__ mask, int* __restrict__ events,
              float* __restrict__ invcnt, float* __restrict__ padout) {
  __shared__ int lbeg[NID], lend[NID];
  __shared__ int evl[NEV];
  const int t = threadIdx.x;
  const int b = blockIdx.x;

  if (t < NID) { lbeg[t] = SS; lend[t] = -1; }
  __syncthreads();

  for (int s = t; s < SS; s += 256) {
    int m = mask[b * SS + s];
    if (m >= 0 && m < NID) {
      atomicMin(&lbeg[m], s);
      atomicMax(&lend[m], s);
    }
  }
  __syncthreads();

  if (t < NID) {
    int e  = lend[t];
    bool ex = (e >= 0);
    int beg = ex ? lbeg[t] : 0;     // empty span at 0 for missing ids -> exact 0
    int end = ex ? (e + 1) : 0;
    int cnt = ex ? (end - beg) : 1;
    invcnt[b * NID + t] = 1.0f / (float)cnt;
    padout[b * NID + t] = ex ? 0.0f : 1.0f;
    evl[t]       = (beg << 8) | (t << 1);        // beg event (weight -1)
    evl[NID + t] = (end << 8) | (t << 1) | 1;    // end event (weight +1)
  }
  __syncthreads();

  // parallel rank sort of 128 packed events (all values distinct in low bits)
  if (t < NEV) {
    int my = evl[t];
    int rank = 0;
#pragma unroll 8
    for (int k = 0; k < NEV; ++k) {
      int o = evl[k];
      rank += (o < my) || (o == my && k < t);
    }
    events[b * NEV + rank] = my;
  }
}

// ---------------------------------------------------------------------------
// Kernel A: stream hidden through LDS (async double-buffer), record
//   - chunk totals C[b][j][h]
//   - local exclusive prefix r_evt[b][k][h] at every event inside the chunk
// ---------------------------------------------------------------------------
__global__ __launch_bounds__(256)
void kA_stream(const float* __restrict__ hidden, const int* __restrict__ events,
               float* __restrict__ Cout, float* __restrict__ revt) {
  __shared__ float buf[2][RB * HT];   // 2 x 16KB
  __shared__ int   ev[NEV];

  const int t  = threadIdx.x;
  const int hx = blockIdx.x;          // 0..7
  const int j  = blockIdx.y;          // 0..15 (s-chunk)
  const int b  = blockIdx.z;          // 0..15
  const int h0 = hx * HT;
  const int s0 = j * CS;

  if (t < NEV) ev[t] = events[b * NEV + t];
  __syncthreads();

  // lower bound of events belonging to this chunk (uniform scalar scan)
  int ptr = 0;
  while (ptr < NEV && (ev[ptr] >> 8) < s0) ++ptr;

  const float* gbase = hidden + (size_t)b * SS * HH + h0;
  unsigned lb0 = (unsigned)(size_t)(&buf[0][0]);
  unsigned lb1 = (unsigned)(size_t)(&buf[1][0]);

  float rsum = 0.0f;

  async_fill_rows(gbase + (size_t)s0 * HH, lb0, t);   // prime pipeline

  for (int st = 0; st < NSTG; ++st) {
    const int cur = st & 1;
    if (st + 1 < NSTG) {
      async_fill_rows(gbase + (size_t)(s0 + (st + 1) * RB) * HH,
                      (cur ? lb0 : lb1), t);
      wait_async_le4();
    } else {
      wait_async_0();
    }
    __syncthreads();   // all waves' fills for `cur` complete

#pragma unroll
    for (int r = 0; r < RB; ++r) {
      const int s = s0 + st * RB + r;
      while (ptr < NEV && (ev[ptr] >> 8) == s) {   // exclusive prefix at s
        revt[(size_t)(b * NEV + ptr) * HH + h0 + t] = rsum;
        ++ptr;
      }
      rsum += buf[cur][r * HT + t];
    }
    __syncthreads();   // done reading `cur` before it gets refilled
  }

  Cout[(size_t)(b * NS + j) * HH + h0 + t] = rsum;
}

// ---------------------------------------------------------------------------
// Kernel B: gathered[n][h] = sum_k M[n][k] * P[k][h], via V_WMMA_F32_16X16X4_F32
//   M: 64x128 (+1 end / -1 beg / 0), P: 128xHB prefix values at event positions
// ---------------------------------------------------------------------------
__global__ __launch_bounds__(256)
void kB_wmma(const float* __restrict__ Cin, const float* __restrict__ revt,
             const int* __restrict__ events, const float* __restrict__ invcnt,
             float* __restrict__ out) {
  __shared__ float Mlds[NID * MP];     // 64x129 padded  (~33KB)
  __shared__ float Pm[NEV * HB];       // 128x128        (64KB)
  __shared__ float preL[NS * HB];      // 16x128         (8KB)
  __shared__ float totalL[HB];
  __shared__ int   ev[NEV];
  __shared__ float ic[NID];

  const int t  = threadIdx.x;
  const int hx = blockIdx.x;
  const int b  = blockIdx.y;
  const int h0 = hx * HB;

  if (t < NEV) ev[t] = events[b * NEV + t];
  if (t < NID) ic[t] = invcnt[b * NID + t];
#pragma unroll
  for (int i = 0; i < (NID * MP + 255) / 256; ++i) {
    int idx = t + 256 * i;
    if (idx < NID * MP) Mlds[idx] = 0.0f;
  }
  __syncthreads();

  // Build M (+1 / -1 at event columns) and per-column chunk prefixes
  if (t < NEV) {
    int e = ev[t];
    int n = (e >> 1) & 63;
    Mlds[n * MP + t] = (e & 1) ? 1.0f : -1.0f;
  }
  if (t < HB) {
    float run = 0.0f;
#pragma unroll
    for (int jj = 0; jj < NS; ++jj) {
      preL[jj * HB + t] = run;
      run += Cin[(size_t)(b * NS + jj) * HH + h0 + t];
    }
    totalL[t] = run;
  }
  __syncthreads();

  // P[k][col] = global exclusive prefix at event k's position
  {
    const int col  = t & (HB - 1);
    const int half = t >> 7;            // 0 or 1 -> events [0,64) / [64,128)
    for (int k = half * 64; k < half * 64 + 64; ++k) {
      int e = ev[k];
      int p = e >> 8;
      float P = (p >= SS) ? totalL[col]
              : preL[(p / CS) * HB + col]
                + revt[(size_t)(b * NEV + k) * HH + h0 + col];
      Pm[k * HB + col] = P;
    }
  }
  __syncthreads();

  // WMMA tile loop: wave w handles n-tile w; 4 m-tiles x 32 k-steps
  const int wv   = t >> 5;              // 0..7
  const int lane = t & 31;
  const int lcol = lane & 15;
  const int klo  = (lane < 16) ? 0 : 2;
  const int n0   = wv * 16;

  v8f acc0 = {}, acc1 = {}, acc2 = {}, acc3 = {};
  for (int kt = 0; kt < 32; ++kt) {
    const int kb = kt * 4 + klo;
    v2f bv;
    bv.x = Pm[(kb + 0) * HB + n0 + lcol];
    bv.y = Pm[(kb + 1) * HB + n0 + lcol];
#pragma unroll
    for (int mt = 0; mt < 4; ++mt) {
      v2f av;
      av.x = Mlds[(mt * 16 + lcol) * MP + kb + 0];
      av.y = Mlds[(mt * 16 + lcol) * MP + kb + 1];
      v8f* accp = (mt == 0) ? &acc0 : (mt == 1) ? &acc1 : (mt == 2) ? &acc2 : &acc3;
      *accp = __builtin_amdgcn_wmma_f32_16x16x4_f32(
          /*neg_a=*/false, av, /*neg_b=*/false, bv,
          /*c_mod=*/(short)0, *accp, /*reuse_a=*/false, /*reuse_b=*/false);
    }
  }

  // Write D tiles: lane l, VGPR v -> row = v + 8*(l>=16), col = l&15
  const int rhi = (lane >> 4) * 8;
#pragma unroll
  for (int mt = 0; mt < 4; ++mt) {
    const v8f acc = (mt == 0) ? acc0 : (mt == 1) ? acc1 : (mt == 2) ? acc2 : acc3;
#pragma unroll
    for (int v = 0; v < 8; ++v) {
      const int nid = mt * 16 + v + rhi;
      out[(size_t)(b * NID + nid) * HH + h0 + n0 + lcol] = acc[v] * ic[nid];
    }
  }
}

// ---------------------------------------------------------------------------
// Launch
// ---------------------------------------------------------------------------
extern "C" void kernel_launch(void* const* d_in, const int* in_sizes, int n_in,
                              void* d_out, int out_size, void* d_ws, size_t ws_size,
                              hipStream_t stream) {
  const float* hidden = (const float*)d_in[0];
  const int*   mask   = (const int*)d_in[1];
  // d_in[2] = max_numbers (static 64, unused)

  float* out = (float*)d_out;                 // [B*N*H] gathered ++ [B*N] pad_mask
  float* pad = out + (size_t)BB * NID * HH;

  char* ws = (char*)d_ws;
  int*   events = (int*)ws;                                        // B*128 ints
  float* invcnt = (float*)(ws + (size_t)BB * NEV * 4);             // B*64
  float* C      = (float*)(ws + (size_t)BB * NEV * 4
                              + (size_t)BB * NID * 4);             // B*NS*H
  float* revt   = (float*)(ws + (size_t)BB * NEV * 4
                              + (size_t)BB * NID * 4
                              + (size_t)BB * NS * HH * 4);         // B*128*H

  k0_spans<<<BB, 256, 0, stream>>>(mask, events, invcnt, pad);

  dim3 gA(HH / HT, NS, BB);                   // (8, 16, 16) = 2048 blocks
  kA_stream<<<gA, 256, 0, stream>>>(hidden, events, C, revt);

  dim3 gB(HH / HB, BB);                       // (16, 16) = 256 blocks
  kB_wmma<<<gB, 256, 0, stream>>>(C, revt, events, invcnt, out);
}